// GRUModel_1786706395316
// MI455X (gfx1250) — compile-verified
//
#include <hip/hip_runtime.h>
#include <math.h>

// GRU stack on MI455X (gfx1250, wave32).
//  B=512, T=1024, H=64, G=3H=192.  Layer 0 input size 5, layers 1..4 input 64.
//  One kernel per layer: each workgroup owns a 16-row batch tile and runs the
//  full time recurrence with h resident in LDS (recurrence is independent per
//  batch row -> no inter-block sync).  Matmuls use V_WMMA_F32_16X16X4_F32
//  (full fp32 precision for the 1024-step recurrence).
//  Per step: 2 barriers; next-step inputs are loaded into registers at the top
//  of the step (latency hidden behind the WMMA chains) and committed to a
//  double-buffered LDS tile after the GEMM barrier.

typedef float v2f __attribute__((ext_vector_type(2)));
typedef float v8f __attribute__((ext_vector_type(8)));

#define BSZ 512
#define TSTEPS 1024
#define HID 64
#define GATES 192     // 3*HID
#define BT 16         // batch tile per workgroup
#define NTHREADS 768  // 24 waves: 12 gh-GEMM waves + 12 xg waves

__device__ __forceinline__ float sigm(float x) {
    return 1.0f / (1.0f + __expf(-x));
}
__device__ __forceinline__ float fast_tanh(float x) {
    return 2.0f * sigm(2.0f * x) - 1.0f;  // one v_exp_f32
}

// One GRU layer: scan over time, fused input-side + hidden-side GEMMs.
// IN = 5 (layer 0, VALU input path) or 64 (layers 1..4, WMMA input path).
template <int IN>
__global__ __launch_bounds__(NTHREADS)
void gru_scan(const float* __restrict__ xin,   // [B, T, IN]
              const float* __restrict__ w_ih,  // [G, IN]
              const float* __restrict__ w_hh,  // [G, H]
              const float* __restrict__ b_ih,  // [G]
              const float* __restrict__ b_hh,  // [G]
              float* __restrict__ hout)        // [B, T, H]
{
    // Odd-padded rows keep the column-major A-fragment reads conflict-free.
    __shared__ float hs[BT][HID + 1];        // h_{t-1} tile
    __shared__ float xs[2][BT][IN + 1];      // double-buffered x_t tile
    __shared__ float gh[BT][GATES + 1];      // h @ w_hh^T + b_hh
    __shared__ float xg[BT][GATES + 1];      // x @ w_ih^T + b_ih
    __shared__ float wih0[(IN == 5) ? GATES * IN : 1];  // layer-0 weight stash
    __shared__ float bih0[(IN == 5) ? GATES : 1];

    const int tid  = threadIdx.x;
    const int wave = tid >> 5;            // 0..23
    const int lane = tid & 31;
    const int col  = lane & 15;           // tile row (A) / tile col (B, C/D)
    const int half = lane >> 4;           // 0 or 1
    const int b0   = blockIdx.x * BT;

    // h_0 = 0
    for (int e = tid; e < BT * (HID + 1); e += NTHREADS)
        (&hs[0][0])[e] = 0.0f;
    if (IN == 5) {  // stash tiny layer-0 input weights in LDS once
        for (int e = tid; e < GATES * IN; e += NTHREADS) wih0[e] = w_ih[e];
        for (int e = tid; e < GATES; e += NTHREADS)      bih0[e] = b_ih[e];
    }

    const bool is_gh = (wave < 12);
    const int  nt    = is_gh ? wave : (wave - 12);  // which 16-wide gate tile
    const int  gcol  = nt * 16 + col;               // global gate column [0,192)

    // ---- Preload B-fragments (weights are time-invariant) -----------------
    // gh = h @ w_hh^T  =>  B[k][n] = w_hh[gcol][k]   (K = 64 -> 16 chunks)
    // xg = x @ w_ih^T  =>  B[k][n] = w_ih[gcol][k]   (only when IN == HID)
    // Layout per ISA 7.12.2: v.x holds K = 4*kc + 2*half, v.y holds K+1.
    v2f bfrag[16];
    if (is_gh) {
#pragma unroll
        for (int kc = 0; kc < 16; ++kc) {
            const int k = kc * 4 + half * 2;
            bfrag[kc].x = w_hh[gcol * HID + k];
            bfrag[kc].y = w_hh[gcol * HID + k + 1];
        }
    } else if (IN == HID) {
#pragma unroll
        for (int kc = 0; kc < 16; ++kc) {
            const int k = kc * 4 + half * 2;
            bfrag[kc].x = w_ih[gcol * IN + k];
            bfrag[kc].y = w_ih[gcol * IN + k + 1];
        }
    }
    const float bias = is_gh ? b_hh[gcol] : b_ih[gcol];

    // ---- prologue: stage x_0 ----------------------------------------------
    for (int e = tid; e < BT * IN; e += NTHREADS) {
        const int r = e / IN, c = e % IN;
        xs[0][r][c] = xin[((size_t)(b0 + r) * TSTEPS) * IN + c];
    }
    __syncthreads();

    constexpr int NX = (BT * IN + NTHREADS - 1) / NTHREADS;  // per-thread stage elems

    for (int t = 0; t < TSTEPS; ++t) {
        const int buf = t & 1;

        // ---- issue next-step loads into registers; overlap with the GEMMs -
        float xreg[NX];
        if (t + 1 < TSTEPS) {
#pragma unroll
            for (int i = 0; i < NX; ++i) {
                const int e = tid + i * NTHREADS;
                if (e < BT * IN) {
                    const int r = e / IN, c = e % IN;
                    xreg[i] = xin[((size_t)(b0 + r) * TSTEPS + (t + 1)) * IN + c];
                }
            }
            if (t + 2 < TSTEPS && tid < BT)  // nudge t+2 toward L2
                __builtin_prefetch(&xin[((size_t)(b0 + tid) * TSTEPS + (t + 2)) * IN], 0, 0);
        }

        // ---- the two 16x192 GEMMs, one 16x16 tile per wave ----------------
        // Two interleaved accumulator chains halve the WMMA serial latency;
        // bias is folded into the acc0 initial value (one column per lane).
        if (is_gh) {
            v8f acc0 = {bias, bias, bias, bias, bias, bias, bias, bias};
            v8f acc1 = {};
#pragma unroll
            for (int kc = 0; kc < 16; kc += 2) {
                const int k = kc * 4 + half * 2;
                v2f a0, a1;
                a0.x = hs[col][k];     a0.y = hs[col][k + 1];
                a1.x = hs[col][k + 4]; a1.y = hs[col][k + 5];
                acc0 = __builtin_amdgcn_wmma_f32_16x16x4_f32(
                    false, a0, false, bfrag[kc], (short)0, acc0, false, false);
                acc1 = __builtin_amdgcn_wmma_f32_16x16x4_f32(
                    false, a1, false, bfrag[kc + 1], (short)0, acc1, false, false);
            }
#pragma unroll
            for (int r = 0; r < 8; ++r)
                gh[r + half * 8][gcol] = acc0[r] + acc1[r];
        } else if (IN == HID) {
            v8f acc0 = {bias, bias, bias, bias, bias, bias, bias, bias};
            v8f acc1 = {};
#pragma unroll
            for (int kc = 0; kc < 16; kc += 2) {
                const int k = kc * 4 + half * 2;
                v2f a0, a1;
                a0.x = xs[buf][col][k];     a0.y = xs[buf][col][k + 1];
                a1.x = xs[buf][col][k + 4]; a1.y = xs[buf][col][k + 5];
                acc0 = __builtin_amdgcn_wmma_f32_16x16x4_f32(
                    false, a0, false, bfrag[kc], (short)0, acc0, false, false);
                acc1 = __builtin_amdgcn_wmma_f32_16x16x4_f32(
                    false, a1, false, bfrag[kc + 1], (short)0, acc1, false, false);
            }
#pragma unroll
            for (int r = 0; r < 8; ++r)
                xg[r + half * 8][gcol] = acc0[r] + acc1[r];
        } else {
            // IN == 5: tiny input GEMM on VALU from LDS-stashed weights.
            for (int e = lane; e < BT * 16; e += 32) {
                const int r = e >> 4, c = e & 15;
                const int gc = nt * 16 + c;
                float s = bih0[gc];
#pragma unroll
                for (int i = 0; i < IN; ++i)
                    s += xs[buf][r][i] * wih0[gc * IN + i];
                xg[r][gc] = s;
            }
        }
        __syncthreads();  // gh/xg ready; xs[buf^1] no longer being read

        // ---- commit next-step inputs to the alternate LDS buffer ----------
        if (t + 1 < TSTEPS) {
#pragma unroll
            for (int i = 0; i < NX; ++i) {
                const int e = tid + i * NTHREADS;
                if (e < BT * IN) xs[buf ^ 1][e / IN][e % IN] = xreg[i];
            }
        }

        // ---- gates + h update: 16*64 = 1024 elements, one owner each ------
        for (int e = tid; e < BT * HID; e += NTHREADS) {
            const int m = e >> 6, n = e & 63;
            const float r  = sigm(xg[m][n] + gh[m][n]);
            const float z  = sigm(xg[m][HID + n] + gh[m][HID + n]);
            const float ng = fast_tanh(xg[m][2 * HID + n] + r * gh[m][2 * HID + n]);
            const float hn = (1.0f - z) * ng + z * hs[m][n];
            hs[m][n] = hn;
            hout[((size_t)(b0 + m) * TSTEPS + t) * HID + n] = hn;
        }
        __syncthreads();
    }
}

// Final FC on the last timestep: out[b] = h[b, T-1, :] . w_fc + b_fc
__global__ __launch_bounds__(256)
void gru_fc(const float* __restrict__ hbuf, const float* __restrict__ w_fc,
            const float* __restrict__ b_fc, float* __restrict__ out)
{
    const int b = blockIdx.x * blockDim.x + threadIdx.x;
    if (b >= BSZ) return;
    const float* hp = hbuf + ((size_t)b * TSTEPS + (TSTEPS - 1)) * HID;
    float s = b_fc[0];
#pragma unroll
    for (int k = 0; k < HID; ++k) s += hp[k] * w_fc[k];
    out[b] = s;
}

extern "C" void kernel_launch(void* const* d_in, const int* in_sizes, int n_in,
                              void* d_out, int out_size, void* d_ws, size_t ws_size,
                              hipStream_t stream) {
    (void)in_sizes; (void)n_in; (void)out_size; (void)ws_size;
    const float* x     = (const float*)d_in[0];   // [512,1024,5]
    const float* w_ih0 = (const float*)d_in[1];   // [192,5]
    const float* w_hh0 = (const float*)d_in[2];   // [192,64]
    const float* b_ih0 = (const float*)d_in[3];   // [192]
    const float* b_hh0 = (const float*)d_in[4];   // [192]
    const float* w_ih  = (const float*)d_in[5];   // [4,192,64]
    const float* w_hh  = (const float*)d_in[6];   // [4,192,64]
    const float* b_ih  = (const float*)d_in[7];   // [4,192]
    const float* b_hh  = (const float*)d_in[8];   // [4,192]
    const float* w_fc  = (const float*)d_in[9];   // [1,64]
    const float* b_fc  = (const float*)d_in[10];  // [1]

    // Workspace: two ping-pong hidden-state buffers, 134 MB each.
    const size_t HBUF = (size_t)BSZ * TSTEPS * HID;
    float* buf0 = (float*)d_ws;
    float* buf1 = buf0 + HBUF;

    dim3 grid(BSZ / BT), block(NTHREADS);

    // Layer 0 (input size 5)
    gru_scan<5><<<grid, block, 0, stream>>>(x, w_ih0, w_hh0, b_ih0, b_hh0, buf0);

    // Layers 1..4 (input size 64)
    float* src = buf0;
    float* dst = buf1;
    for (int l = 0; l < 4; ++l) {
        gru_scan<64><<<grid, block, 0, stream>>>(
            src,
            w_ih + (size_t)l * GATES * HID,
            w_hh + (size_t)l * GATES * HID,
            b_ih + (size_t)l * GATES,
            b_hh + (size_t)l * GATES,
            dst);
        float* tmp = src; src = dst; dst = tmp;
    }
    // After 5 layers the final states are in `src` (== buf0).
    gru_fc<<<(BSZ + 255) / 256, 256, 0, stream>>>(src, w_fc, b_fc, (float*)d_out);
}